// StatisticalFeatures_59725815218222
// MI455X (gfx1250) — compile-verified
//
#include <hip/hip_runtime.h>
#include <math.h>

// Sizes from the reference
#define Bn 32
#define Ln 4096
#define Nn 256
#define Fb 2049          // rfft bins
#define KTOP 5
#define BLK3 128

typedef float v2f __attribute__((ext_vector_type(2)));
typedef float v8f __attribute__((ext_vector_type(8)));

// One K=4 slice of a 16x16xK f32 WMMA accumulate chain.
// A (16x4 f32): lanes 0-15 rows, v0={K0|K2}, v1={K1|K3} split by lane-half.
// B (4x16 f32): same K split pattern, col = lane&15. D/C: v8f per ISA layout.
__device__ __forceinline__ v8f wmma_f32_k4(float a0, float a1, float b0, float b1, v8f c) {
  v2f a = {a0, a1};
  v2f b = {b0, b1};
  return __builtin_amdgcn_wmma_f32_16x16x4_f32(false, a, false, b, (short)0, c, false, false);
}

// ---------------------------------------------------------------------------
// Kernel 1: streaming raw sums S1..S4 and lag cross-products (lags 1,3,6,12,24)
// grid (B, 16 chunks of 256 timesteps), block 256 (one thread per channel n).
// Coalesced loads across n; register shift-window for lag history; atomics to ws.
// ---------------------------------------------------------------------------
__global__ void __launch_bounds__(256)
k_moments(const float* __restrict__ x, float* __restrict__ sums) {
  const int b = blockIdx.x;
  const int chunk = blockIdx.y;
  const int n = threadIdx.x;
  const float* xb = x + (size_t)b * Ln * Nn + n;
  const int t0 = chunk * 256;

  float h[24];  // h[j] = x[t_group_start - 24 + j]  (0 before t=0)
#pragma unroll
  for (int j = 0; j < 24; ++j) {
    int t = t0 - 24 + j;
    h[j] = (t >= 0) ? xb[(size_t)t * Nn] : 0.0f;
  }
  float s1 = 0.f, s2 = 0.f, s3 = 0.f, s4 = 0.f;
  float p1 = 0.f, p3 = 0.f, p6 = 0.f, p12 = 0.f, p24 = 0.f;

  for (int g = 0; g < 256; g += 8) {
    float gv[8];
#pragma unroll
    for (int i = 0; i < 8; ++i) gv[i] = xb[(size_t)(t0 + g + i) * Nn];
    if (g + 8 < 256) __builtin_prefetch(xb + (size_t)(t0 + g + 8) * Nn, 0, 1);
#pragma unroll
    for (int i = 0; i < 8; ++i) {
      float v = gv[i], v2 = v * v;
      s1 += v; s2 += v2; s3 += v2 * v; s4 += v2 * v2;
      float a1 = (i >= 1) ? gv[(i >= 1) ? (i - 1) : 0] : h[(23 + i) % 24];
      float a3 = (i >= 3) ? gv[(i >= 3) ? (i - 3) : 0] : h[(21 + i) % 24];
      float a6 = (i >= 6) ? gv[(i >= 6) ? (i - 6) : 0] : h[(18 + i) % 24];
      p1  += v * a1;
      p3  += v * a3;
      p6  += v * a6;
      p12 += v * h[12 + i];
      p24 += v * h[i];
    }
#pragma unroll
    for (int j = 0; j < 24; ++j) h[j] = (j < 16) ? h[j + 8] : gv[j - 16];
  }
  float* dst = sums + ((size_t)b * Nn + n) * 9;
  atomicAdd(dst + 0, s1); atomicAdd(dst + 1, s2);
  atomicAdd(dst + 2, s3); atomicAdd(dst + 3, s4);
  atomicAdd(dst + 4, p1); atomicAdd(dst + 5, p3);
  atomicAdd(dst + 6, p6); atomicAdd(dst + 7, p12);
  atomicAdd(dst + 8, p24);
}

// ---------------------------------------------------------------------------
// Kernel 2: finalize mean/std(ddof=1)/skew/kurt and 5 autocorrelations.
// Autocorr from raw sums: sum x_t x_{t+lag} - m*(sumA+sumB) + (L-lag)m^2,
// with head/tail partial sums read directly (<=24 elems each end).
// ---------------------------------------------------------------------------
__global__ void __launch_bounds__(256)
k_finalize(const float* __restrict__ x, const float* __restrict__ sums,
           float* __restrict__ out) {
  const int b = blockIdx.x;
  const int n = threadIdx.x;
  const float eps = 1e-8f;
  const float* s = sums + ((size_t)b * Nn + n) * 9;
  float S1 = s[0], S2 = s[1], S3 = s[2], S4 = s[3];
  float P[5] = {s[4], s[5], s[6], s[7], s[8]};

  const float invL = 1.0f / (float)Ln;
  float m = S1 * invL;
  float var1 = (S2 - (float)Ln * m * m) / (float)(Ln - 1);
  var1 = fmaxf(var1, 0.0f);
  float sd = sqrtf(var1);
  float sde = sd + eps;
  float e2 = S2 * invL, e3 = S3 * invL, e4 = S4 * invL;
  float m3 = e3 - 3.0f * m * e2 + 2.0f * m * m * m;
  float m4 = e4 - 4.0f * m * e3 + 6.0f * m * m * e2 - 3.0f * m * m * m * m;
  float sde2 = sde * sde;
  float skew = m3 / (sde2 * sde + eps);
  float kurt = m4 / (sde2 * sde2 + eps) - 3.0f;

  const float* xb = x + (size_t)b * Ln * Nn + n;
  float head[24], tail[24];
#pragma unroll
  for (int j = 0; j < 24; ++j) head[j] = xb[(size_t)j * Nn];
#pragma unroll
  for (int j = 0; j < 24; ++j) tail[j] = xb[(size_t)(Ln - 24 + j) * Nn];

  float* o = out + ((size_t)b * Nn + n) * 19;
  o[0] = m; o[1] = sd; o[2] = skew; o[3] = kurt;

  const int lags[5] = {1, 3, 6, 12, 24};
#pragma unroll
  for (int li = 0; li < 5; ++li) {
    int lag = lags[li];
    float hs = 0.f, ts = 0.f;
    for (int j = 0; j < lag; ++j) { hs += head[j]; ts += tail[23 - j]; }
    float sumA = S1 - hs;   // sum_{t=lag}^{L-1} x
    float sumB = S1 - ts;   // sum_{t=0}^{L-lag-1} x
    float cnt = (float)(Ln - lag);
    float cov = P[li] - m * sumA - m * sumB + cnt * m * m;
    o[14 + li] = cov / (cnt * sde2);
  }
}

// ---------------------------------------------------------------------------
// Kernel 3: per-(b,n) 4096-pt FFT as three radix-16 WMMA stages (in-place in
// LDS), amplitude, iterative multi-level top-k with gate MLP, final decode.
// block = 128 threads (4 waves, 4 16x16 tiles per wave per stage).
// ---------------------------------------------------------------------------
__global__ void __launch_bounds__(BLK3)
k_fft_topk(const float* __restrict__ x,
           const float* __restrict__ w1, const float* __restrict__ b1v,
           const float* __restrict__ w2, const float* __restrict__ b2v,
           float* __restrict__ out) {
  __shared__ float dre[4096];
  __shared__ float dim_[4096];
  __shared__ float dftre[256];   // Re W16^{f k}
  __shared__ float dftim[256];   // Im W16^{f k} = -sin
  __shared__ float amp[Fb];
  __shared__ float red_v[BLK3];
  __shared__ int   red_i[BLK3];
  __shared__ float pk_v[KTOP];
  __shared__ int   pk_i[KTOP];

  const int n = blockIdx.x;
  const int b = blockIdx.y;
  const int tid = threadIdx.x;
  const int wave = tid >> 5;
  const int mrow = tid & 15;     // lane&15 (lane == tid&31)
  const int hi = (tid >> 4) & 1; // lane>>4

  // Load series (real) into LDS.
  const float* xs = x + (size_t)b * Ln * Nn + n;
  for (int t = tid; t < 4096; t += BLK3) dre[t] = xs[(size_t)t * Nn];
  // DFT-16 matrix table.
  for (int i = tid; i < 256; i += BLK3) {
    int f = i >> 4, k = i & 15;
    float th = (float)((f * k) & 15) * (6.28318530717958647692f / 16.0f);
    dftre[i] = __cosf(th);
    dftim[i] = -__sinf(th);
  }
  __syncthreads();

  const float TWOPI_L = 6.28318530717958647692f / 4096.0f;

  // ---- Stage 1: DFT16 over t2 (real input), fuse twiddle W4096^{16*t1*f0} ----
  for (int c = wave; c < 16; c += 4) {
    const int base = (c << 4) + mrow;
    float br[8];
#pragma unroll
    for (int kk = 0; kk < 4; ++kk) {
      int k0 = kk * 4 + 2 * hi;
      br[2 * kk]     = dre[base + (k0 << 8)];
      br[2 * kk + 1] = dre[base + ((k0 + 1) << 8)];
    }
    v8f cre = {0, 0, 0, 0, 0, 0, 0, 0};
    v8f cim = {0, 0, 0, 0, 0, 0, 0, 0};
#pragma unroll
    for (int kk = 0; kk < 4; ++kk) {
      int k0 = kk * 4 + 2 * hi;
      float c0 = dftre[(mrow << 4) + k0], c1 = dftre[(mrow << 4) + k0 + 1];
      float d0 = dftim[(mrow << 4) + k0], d1 = dftim[(mrow << 4) + k0 + 1];
      cre = wmma_f32_k4(c0, c1, br[2 * kk], br[2 * kk + 1], cre);
      cim = wmma_f32_k4(d0, d1, br[2 * kk], br[2 * kk + 1], cim);
    }
#pragma unroll
    for (int r = 0; r < 8; ++r) {
      int M = r + 8 * hi;                  // f0
      int idx = (M << 8) + base;
      int ti = (16 * c * M) & 4095;        // t1 = c
      float th = TWOPI_L * (float)ti;
      float tc = __cosf(th), ts = -__sinf(th);
      float re = cre[r], im = cim[r];
      dre[idx] = re * tc - im * ts;
      dim_[idx] = re * ts + im * tc;
    }
  }
  __syncthreads();

  // ---- Stage 2: DFT16 over t1 per f0 block, fuse twiddle W4096^{t0*(f0+16f1)} ----
  for (int c = wave; c < 16; c += 4) {       // c = f0
    const int base = (c << 8) + mrow;        // col n = t0 = mrow
    float br[8], bi[8];
#pragma unroll
    for (int kk = 0; kk < 4; ++kk) {
      int k0 = kk * 4 + 2 * hi;              // row k = t1
      br[2 * kk]     = dre[base + (k0 << 4)];
      br[2 * kk + 1] = dre[base + ((k0 + 1) << 4)];
      bi[2 * kk]     = dim_[base + (k0 << 4)];
      bi[2 * kk + 1] = dim_[base + ((k0 + 1) << 4)];
    }
    v8f cre = {0, 0, 0, 0, 0, 0, 0, 0};
    v8f cim = {0, 0, 0, 0, 0, 0, 0, 0};
#pragma unroll
    for (int kk = 0; kk < 4; ++kk) {
      int k0 = kk * 4 + 2 * hi;
      float c0 = dftre[(mrow << 4) + k0], c1 = dftre[(mrow << 4) + k0 + 1];
      float d0 = dftim[(mrow << 4) + k0], d1 = dftim[(mrow << 4) + k0 + 1];
      cre = wmma_f32_k4(c0, c1, br[2 * kk], br[2 * kk + 1], cre);
      cre = wmma_f32_k4(-d0, -d1, bi[2 * kk], bi[2 * kk + 1], cre);
      cim = wmma_f32_k4(d0, d1, br[2 * kk], br[2 * kk + 1], cim);
      cim = wmma_f32_k4(c0, c1, bi[2 * kk], bi[2 * kk + 1], cim);
    }
#pragma unroll
    for (int r = 0; r < 8; ++r) {
      int M = r + 8 * hi;                    // f1
      int idx = (c << 8) + (M << 4) + mrow;  // [f0*256 + f1*16 + t0]
      int ti = (mrow * (c + 16 * M)) & 4095; // t0*(f0+16*f1)
      float th = TWOPI_L * (float)ti;
      float tc = __cosf(th), ts = -__sinf(th);
      float re = cre[r], im = cim[r];
      dre[idx] = re * tc - im * ts;
      dim_[idx] = re * ts + im * tc;
    }
  }
  __syncthreads();

  // ---- Stage 3: DFT16 over t0, write amplitudes for f<2049 ----
  for (int c = wave; c < 16; c += 4) {
    const int base = (c << 8) + (mrow << 4); // pair p = 16c+mrow, + k(t0)
    float br[8], bi[8];
#pragma unroll
    for (int kk = 0; kk < 4; ++kk) {
      int k0 = kk * 4 + 2 * hi;
      br[2 * kk]     = dre[base + k0];
      br[2 * kk + 1] = dre[base + k0 + 1];
      bi[2 * kk]     = dim_[base + k0];
      bi[2 * kk + 1] = dim_[base + k0 + 1];
    }
    v8f cre = {0, 0, 0, 0, 0, 0, 0, 0};
    v8f cim = {0, 0, 0, 0, 0, 0, 0, 0};
#pragma unroll
    for (int kk = 0; kk < 4; ++kk) {
      int k0 = kk * 4 + 2 * hi;
      float c0 = dftre[(mrow << 4) + k0], c1 = dftre[(mrow << 4) + k0 + 1];
      float d0 = dftim[(mrow << 4) + k0], d1 = dftim[(mrow << 4) + k0 + 1];
      cre = wmma_f32_k4(c0, c1, br[2 * kk], br[2 * kk + 1], cre);
      cre = wmma_f32_k4(-d0, -d1, bi[2 * kk], bi[2 * kk + 1], cre);
      cim = wmma_f32_k4(d0, d1, br[2 * kk], br[2 * kk + 1], cim);
      cim = wmma_f32_k4(c0, c1, bi[2 * kk], bi[2 * kk + 1], cim);
    }
#pragma unroll
    for (int r = 0; r < 8; ++r) {
      int M = r + 8 * hi;                    // f2
      int f = c + (mrow << 4) + (M << 8);    // f0 + 16 f1 + 256 f2
      if (f < Fb) amp[f] = sqrtf(cre[r] * cre[r] + cim[r] * cim[r]);
    }
  }
  __syncthreads();

  // ---- Reduction helpers ----
  auto block_sum = [&](const float* arr, int len) -> float {
    float s = 0.f;
    for (int i = tid; i < len; i += BLK3) s += arr[i];
    red_v[tid] = s;
    __syncthreads();
    for (int off = BLK3 / 2; off > 0; off >>= 1) {
      if (tid < off) red_v[tid] += red_v[tid + off];
      __syncthreads();
    }
    float r = red_v[0];
    __syncthreads();
    return r;
  };
  auto block_maxv = [&](const float* arr, int len) -> float {
    float s = -1e30f;
    for (int i = tid; i < len; i += BLK3) s = fmaxf(s, arr[i]);
    red_v[tid] = s;
    __syncthreads();
    for (int off = BLK3 / 2; off > 0; off >>= 1) {
      if (tid < off) red_v[tid] = fmaxf(red_v[tid], red_v[tid + off]);
      __syncthreads();
    }
    float r = red_v[0];
    __syncthreads();
    return r;
  };
  auto block_argmax = [&](const float* arr, int len) {
    float bv = -1e30f; int bix = 0x7fffffff;
    for (int i = tid; i < len; i += BLK3) {
      float v = arr[i];
      if (v > bv) { bv = v; bix = i; }
    }
    red_v[tid] = bv; red_i[tid] = bix;
    __syncthreads();
    for (int off = BLK3 / 2; off > 0; off >>= 1) {
      if (tid < off) {
        float ov = red_v[tid + off]; int oi = red_i[tid + off];
        if (ov > red_v[tid] || (ov == red_v[tid] && oi < red_i[tid])) {
          red_v[tid] = ov; red_i[tid] = oi;
        }
      }
      __syncthreads();
    }
    // result in red_v[0], red_i[0]
  };

  const float eps = 1e-8f;
  float asum = block_sum(amp, Fb);
  float amax = block_maxv(amp, Fb);
  float total_energy = asum / (float)Fb + eps;

  // residual -> dre[0..2048], aggregated -> dim_[0..2048] (FFT buffers free now)
  for (int i = tid; i < Fb; i += BLK3) { dre[i] = amp[i]; dim_[i] = 0.0f; }
  __syncthreads();

  float running = 1.0f;                      // identical on every thread
  const int klev[4] = {5, 2, 1, 1};          // round(5/2^l), banker's rounding
  for (int lvl = 0; lvl < 4; ++lvl) {
    float rsum = block_sum(dre, Fb);         // residual mean BEFORE subtraction
    int kl = klev[lvl];
    for (int j = 0; j < kl; ++j) {
      block_argmax(dre, Fb);
      if (tid == 0) {
        pk_v[j] = red_v[0]; pk_i[j] = red_i[0];
        dre[red_i[0]] = -1.0f;               // mask for subsequent picks
      }
      __syncthreads();
    }
    if (tid == 0) {
      for (int j = 0; j < kl; ++j) {
        dim_[pk_i[j]] += running * pk_v[j];  // aggregated += running*level_map
        dre[pk_i[j]] = 0.0f;                 // residual -= level_map
      }
    }
    // gate MLP (all threads compute the same scalar)
    float r = (rsum / (float)Fb) / total_energy;
    float lr = (float)(lvl + 1) * 0.25f;
    float z = b2v[0];
    for (int j = 0; j < 8; ++j) {
      float hpre = r * w1[j] + lr * w1[8 + j] + b1v[j];
      float g = 0.5f * hpre * (1.0f + erff(hpre * 0.70710678118654752f));
      z += g * w2[j];
    }
    running *= 1.0f / (1.0f + __expf(-z));
    __syncthreads();
  }

  // final hard top-5 of aggregated
  for (int j = 0; j < KTOP; ++j) {
    block_argmax(dim_, Fb);
    if (tid == 0) {
      pk_v[j] = red_v[0]; pk_i[j] = red_i[0];
      dim_[red_i[0]] = -1.0f;
    }
    __syncthreads();
  }
  if (tid == 0) {
    float denom = amax + eps + eps;          // (max+eps)+eps per reference
    float* o = out + ((size_t)b * Nn + n) * 19;
    for (int j = 0; j < KTOP; ++j) {
      o[4 + j] = (float)pk_i[j] / (float)Ln;
      o[9 + j] = pk_v[j] / denom;
    }
  }
}

// ---------------------------------------------------------------------------
extern "C" void kernel_launch(void* const* d_in, const int* in_sizes, int n_in,
                              void* d_out, int out_size, void* d_ws, size_t ws_size,
                              hipStream_t stream) {
  (void)in_sizes; (void)n_in; (void)out_size; (void)ws_size;
  const float* x  = (const float*)d_in[0];
  const float* w1 = (const float*)d_in[1];
  const float* b1 = (const float*)d_in[2];
  const float* w2 = (const float*)d_in[3];
  const float* b2 = (const float*)d_in[4];
  float* out = (float*)d_out;
  float* sums = (float*)d_ws;                 // [B][N][9] partial sums

  hipMemsetAsync(d_ws, 0, (size_t)Bn * Nn * 9 * sizeof(float), stream);
  k_moments<<<dim3(Bn, 16), 256, 0, stream>>>(x, sums);
  k_finalize<<<dim3(Bn), 256, 0, stream>>>(x, sums, out);
  k_fft_topk<<<dim3(Nn, Bn), BLK3, 0, stream>>>(x, w1, b1, w2, b2, out);
}